// DispatchCombineOnlyModel_62878321214343
// MI455X (gfx1250) — compile-verified
//
#include <hip/hip_runtime.h>

typedef float v2f __attribute__((ext_vector_type(2)));
typedef float v4f __attribute__((ext_vector_type(4)));
typedef float v8f __attribute__((ext_vector_type(8)));

#define H_DIM   2048
#define E_NUM   64
#define TOPK    8
#define KC      64        // K-chunk staged in LDS per iteration
#define TOK_WG  64        // tokens per workgroup
#define THREADS 128       // 4 waves (wave32); 16 tokens per wave
#define LSTRIDE 68        // padded LDS row stride (floats): 16B-aligned, bank-spread
#define LOGSTRIDE 65      // padded logits stride: conflict-free per-token scans

__global__ __launch_bounds__(THREADS)
void moe_router_combine_kernel(const float* __restrict__ X,
                               const float* __restrict__ W,
                               const float* __restrict__ RB,
                               const float* __restrict__ EB,
                               float* __restrict__ Out)
{
    __shared__ float xs[TOK_WG * LSTRIDE];   // x tile; reused as logits after GEMM
    __shared__ float ws[E_NUM * LSTRIDE];    // router weight chunk
    __shared__ float tw  [TOK_WG * TOPK];    // top-k weights
    __shared__ int   tix [TOK_WG * TOPK];    // top-k expert indices
    __shared__ float wsum_s[TOK_WG];         // sum of top-k weights

    const int tid  = threadIdx.x;
    const int lane = tid & 31;
    const int wave = tid >> 5;
    const int tokBase = blockIdx.x * TOK_WG;

    // WMMA f32 16x16x4 operand layout: lane = 16*h + m ; VGPR v holds K = 2h+v
    const int m  = lane & 15;
    const int kh = (lane >> 4) << 1;   // 0 or 2

    v8f acc[4] = {};                   // 16 tokens x 64 experts (4 N-tiles)

    // ---------------- Phase 1: router GEMM (fp32 WMMA) ----------------
    for (int k0 = 0; k0 < H_DIM; k0 += KC) {
        // Stage x tile: 64 rows x 64 cols (coalesced float4)
        for (int i = tid; i < TOK_WG * (KC / 4); i += THREADS) {
            int r = i >> 4;
            int c = (i & 15) << 2;
            v4f v = *(const v4f*)(X + (size_t)(tokBase + r) * H_DIM + k0 + c);
            *(v4f*)(xs + r * LSTRIDE + c) = v;
        }
        // Stage router-weight chunk: 64 experts x 64 cols (L2-resident source)
        for (int i = tid; i < E_NUM * (KC / 4); i += THREADS) {
            int r = i >> 4;
            int c = (i & 15) << 2;
            v4f v = *(const v4f*)(W + (size_t)r * H_DIM + k0 + c);
            *(v4f*)(ws + r * LSTRIDE + c) = v;
        }
        __syncthreads();

        const float* xrow = xs + (wave * 16 + m) * LSTRIDE + kh;
        #pragma unroll
        for (int kk = 0; kk < KC; kk += 4) {
            v2f av = *(const v2f*)(xrow + kk);
            #pragma unroll
            for (int j = 0; j < 4; ++j) {
                v2f bv = *(const v2f*)(ws + (j * 16 + m) * LSTRIDE + kk + kh);
                acc[j] = __builtin_amdgcn_wmma_f32_16x16x4_f32(
                    /*neg_a=*/false, av, /*neg_b=*/false, bv,
                    /*c_mod=*/(short)0, acc[j],
                    /*reuse_a=*/false, /*reuse_b=*/false);
            }
        }
        __syncthreads();
    }

    // ---------------- Phase 2: dump logits to LDS ----------------
    float* ls = xs;   // reuse x-tile LDS (64*65 <= 64*68)
    {
        const int nl = lane & 15;
        const int hh = lane >> 4;
        #pragma unroll
        for (int j = 0; j < 4; ++j) {
            #pragma unroll
            for (int r = 0; r < 8; ++r) {
                // C layout: VGPR r, lanes 0-15 -> M=r, lanes 16-31 -> M=r+8
                ls[(wave * 16 + 8 * hh + r) * LOGSTRIDE + j * 16 + nl] = acc[j][r];
            }
        }
    }
    __syncthreads();

    // ---------------- Phase 3: softmax + top-8 (one thread per token) ----------------
    if (tid < TOK_WG) {
        float s[E_NUM];
        float mx = -3.402823466e38f;
        #pragma unroll
        for (int e = 0; e < E_NUM; ++e) {
            float v = ls[tid * LOGSTRIDE + e] + RB[e];
            s[e] = v;
            mx = fmaxf(mx, v);
        }
        float sum = 0.f;
        #pragma unroll
        for (int e = 0; e < E_NUM; ++e) {
            float v = __expf(s[e] - mx);
            s[e] = v;
            sum += v;
        }
        const float inv = 1.f / sum;
        unsigned long long used = 0ull;
        float wacc = 0.f;
        for (int k = 0; k < TOPK; ++k) {
            float best = -1.f;
            int   bi   = 0;
            #pragma unroll
            for (int e = 0; e < E_NUM; ++e) {
                // strict '>' keeps first occurrence on ties, matching lax.top_k
                if ((((used >> e) & 1ull) == 0ull) && (s[e] > best)) { best = s[e]; bi = e; }
            }
            used |= (1ull << bi);
            float wk = best * inv;
            tw [tid * TOPK + k] = wk;
            tix[tid * TOPK + k] = bi;
            wacc += wk;
        }
        wsum_s[tid] = wacc;
    }
    __syncthreads();

    // ---------------- Phase 4: combine (x re-read hits L2) ----------------
    #pragma unroll 1
    for (int t = 0; t < 16; ++t) {
        const int row = wave * 16 + t;
        const size_t token = (size_t)(tokBase + row);
        const float wsum = wsum_s[row];
        float wk[TOPK]; int ik[TOPK];
        #pragma unroll
        for (int k = 0; k < TOPK; ++k) {
            wk[k] = tw [row * TOPK + k];
            ik[k] = tix[row * TOPK + k];
        }
        const float* xr = X   + token * H_DIM;
        float*       orw = Out + token * H_DIM;
        for (int h = lane << 2; h < H_DIM; h += 128) {
            v4f xv = *(const v4f*)(xr + h);
            v4f a  = xv * wsum;
            #pragma unroll
            for (int k = 0; k < TOPK; ++k) {
                v4f bv = *(const v4f*)(EB + (size_t)ik[k] * H_DIM + h);
                a += bv * wk[k];
            }
            *(v4f*)(orw + h) = a;
        }
    }
}

extern "C" void kernel_launch(void* const* d_in, const int* in_sizes, int n_in,
                              void* d_out, int out_size, void* d_ws, size_t ws_size,
                              hipStream_t stream)
{
    (void)n_in; (void)d_ws; (void)ws_size; (void)out_size;
    const float* X  = (const float*)d_in[0];   // hidden_states [4,4096,2048]
    const float* W  = (const float*)d_in[1];   // router_weight [64,2048]
    const float* RB = (const float*)d_in[2];   // router_bias   [64]
    const float* EB = (const float*)d_in[3];   // expert_bias   [64,2048]
    float* Out = (float*)d_out;

    const int T = in_sizes[0] / H_DIM;         // 16384 tokens
    dim3 grid(T / TOK_WG);                     // 256 workgroups
    moe_router_combine_kernel<<<grid, THREADS, 0, stream>>>(X, W, RB, EB, Out);
}